// AttentionSortNet_26465588478196
// MI455X (gfx1250) — compile-verified
//
#include <hip/hip_runtime.h>
#include <math.h>

// Problem constants (from reference)
#define NBH      32
#define SEQ      8192
#define DIMK     64
#define NBUCK    64     // 8192 / 128
#define BSZ      128
#define LPAD     65     // padded LDS row stride (floats) -> conflict-free rows AND cols

typedef __attribute__((ext_vector_type(2))) float v2f;   // fp32 A/B fragment (16x4 / 4x16)
typedef __attribute__((ext_vector_type(8))) float v8f;   // fp32 C/D fragment (16x16)

// ---------------------------------------------------------------------------
// Phase 1: bucket means. One block per (tensor, bh, bucket) = 4096 blocks.
// Each block streams a contiguous 128x64 fp32 tile (32 KB) with float4 loads,
// reduces over rows, writes the 64-dim mean. Pure HBM-bandwidth streaming.
// ---------------------------------------------------------------------------
__global__ __launch_bounds__(256) void bucket_mean_kernel(
    const float* __restrict__ q, const float* __restrict__ k,
    float* __restrict__ ws_sq, float* __restrict__ ws_sk)
{
    __shared__ float4 red[256];

    const int bx     = blockIdx.x;        // 0..4095
    const int tensor = bx >> 11;          // 0 = q, 1 = k
    const int rem    = bx & 2047;
    const int bh     = rem >> 6;
    const int bucket = rem & 63;

    const float* src = tensor ? k : q;
    float*       dst = tensor ? ws_sk : ws_sq;

    const float* base = src + ((size_t)bh * SEQ + (size_t)bucket * BSZ) * DIMK;

    const int t    = threadIdx.x;
    const int col4 = t & 15;   // covers dims [4*col4, 4*col4+4)
    const int rgrp = t >> 4;   // 0..15 row groups

    float4 acc = make_float4(0.f, 0.f, 0.f, 0.f);
    #pragma unroll
    for (int i = 0; i < 8; ++i) {
        const int row = rgrp + (i << 4);
        const float4 v = *reinterpret_cast<const float4*>(base + row * DIMK + (col4 << 2));
        acc.x += v.x; acc.y += v.y; acc.z += v.z; acc.w += v.w;
    }
    red[t] = acc;
    __syncthreads();
    // tree-reduce over the 16 row groups
    for (int s = 8; s > 0; s >>= 1) {
        if (rgrp < s) {
            const float4 o = red[(rgrp + s) * 16 + col4];
            acc.x += o.x; acc.y += o.y; acc.z += o.z; acc.w += o.w;
            red[t] = acc;
        }
        __syncthreads();
    }
    if (rgrp == 0) {
        const float inv = 1.0f / 128.0f;
        float* out = dst + ((size_t)bh * NBUCK + bucket) * DIMK + (col4 << 2);
        out[0] = acc.x * inv; out[1] = acc.y * inv;
        out[2] = acc.z * inv; out[3] = acc.w * inv;
    }
}

// ---------------------------------------------------------------------------
// Phase 2: per batch-head (32 blocks, 4 waves each):
//   R = sq . sk^T * 0.125 via V_WMMA_F32_16X16X4_F32 (fp32-exact path),
//   fused relu/log/gumbel epilogue, 8 Sinkhorn iterations in LDS, exp, store.
// ---------------------------------------------------------------------------
__global__ __launch_bounds__(128) void sortnet_kernel(
    const float* __restrict__ ws_sq, const float* __restrict__ ws_sk,
    const float* __restrict__ ug, float* __restrict__ out)
{
    __shared__ float sqs[64 * LPAD];
    __shared__ float sks[64 * LPAD];
    __shared__ float rmat[64 * LPAD];

    const int bh = blockIdx.x;
    const int t  = threadIdx.x;

    const float* gq  = ws_sq + (size_t)bh * 64 * 64;
    const float* gk  = ws_sk + (size_t)bh * 64 * 64;
    const float* ugb = ug    + (size_t)bh * 64 * 64;

    for (int idx = t; idx < 4096; idx += 128) {
        const int r = idx >> 6, c = idx & 63;
        sqs[r * LPAD + c] = gq[idx];
        sks[r * LPAD + c] = gk[idx];
    }
    __syncthreads();

    // --- GEMM: wave w owns M-tile w (rows 16w..16w+15), all 4 N-tiles ---
    const int lane  = t & 31;
    const int mtile = t >> 5;                    // wave id, 0..3
    const int lrow  = lane & 15;
    const int koff  = (lane >> 4) << 1;          // 0 or 2 (ISA 16x4 fp32 A layout)
    const int mrow  = (mtile << 4) + lrow;
    const int mhi   = (lane >> 4) << 3;          // C layout: lanes 16-31 hold M=v+8

    for (int ntile = 0; ntile < 4; ++ntile) {
        const int nrow = (ntile << 4) + lrow;
        v8f acc = {};
        #pragma unroll
        for (int k0 = 0; k0 < 64; k0 += 4) {
            v2f a, b;
            a.x = sqs[mrow * LPAD + k0 + koff];
            a.y = sqs[mrow * LPAD + k0 + koff + 1];
            b.x = sks[nrow * LPAD + k0 + koff];   // B = sk^T : B[k][n] = sk[n][k]
            b.y = sks[nrow * LPAD + k0 + koff + 1];
            acc = __builtin_amdgcn_wmma_f32_16x16x4_f32(
                false, a, false, b, (short)0, acc, false, false);
        }
        // fused epilogue: R*dim^-0.5 -> log(relu+eps) -> +gumbel -> /T
        const int n = (ntile << 4) + lrow;
        #pragma unroll
        for (int v = 0; v < 8; ++v) {
            const int m = (mtile << 4) + mhi + v;
            const float R = acc[v] * 0.125f;                       // 64^-0.5
            const float u = ugb[m * 64 + n];
            const float gum = -logf(-logf(u + 1e-6f) + 1e-6f);     // -log(-log(u+eps)+eps)
            const float rv  = (logf(fmaxf(R, 0.f) + 1e-6f) + gum) * (1.0f / 0.7f);
            rmat[m * LPAD + n] = rv;
        }
    }
    __syncthreads();

    // --- 8 Sinkhorn iterations: row LSE (axis=2) then col LSE (axis=1) ---
    for (int it = 0; it < 8; ++it) {
        if (t < 64) {                             // row pass: thread t owns row t
            float mx = -INFINITY;
            for (int j = 0; j < 64; ++j) mx = fmaxf(mx, rmat[t * LPAD + j]);
            float s = 0.f;
            for (int j = 0; j < 64; ++j) s += expf(rmat[t * LPAD + j] - mx);
            const float lse = mx + logf(s);
            for (int j = 0; j < 64; ++j) rmat[t * LPAD + j] -= lse;
        }
        __syncthreads();
        if (t < 64) {                             // col pass: thread t owns column t
            float mx = -INFINITY;
            for (int i = 0; i < 64; ++i) mx = fmaxf(mx, rmat[i * LPAD + t]);
            float s = 0.f;
            for (int i = 0; i < 64; ++i) s += expf(rmat[i * LPAD + t] - mx);
            const float lse = mx + logf(s);
            for (int i = 0; i < 64; ++i) rmat[i * LPAD + t] -= lse;
        }
        __syncthreads();
    }

    float* ob = out + (size_t)bh * 64 * 64;
    for (int idx = t; idx < 4096; idx += 128) {
        const int r = idx >> 6, c = idx & 63;
        ob[idx] = expf(rmat[r * LPAD + c]);
    }
}

// ---------------------------------------------------------------------------
extern "C" void kernel_launch(void* const* d_in, const int* in_sizes, int n_in,
                              void* d_out, int out_size, void* d_ws, size_t ws_size,
                              hipStream_t stream)
{
    const float* q  = (const float*)d_in[0];   // [32, 8192, 64] fp32
    const float* k  = (const float*)d_in[1];   // [32, 8192, 64] fp32
    const float* ug = (const float*)d_in[2];   // [32, 64, 64]   fp32
    // d_in[3] = topk (unused)

    float* ws_sq = (float*)d_ws;               // [32, 64, 64] = 512 KB
    float* ws_sk = ws_sq + NBH * NBUCK * DIMK; // [32, 64, 64] = 512 KB

    bucket_mean_kernel<<<dim3(2 * NBH * NBUCK), dim3(256), 0, stream>>>(q, k, ws_sq, ws_sk);
    sortnet_kernel<<<dim3(NBH), dim3(128), 0, stream>>>(ws_sq, ws_sk, ug, (float*)d_out);
}